// MultiResHashGrid_33397665693997
// MI455X (gfx1250) — compile-verified
//
#include <hip/hip_runtime.h>
#include <cstdint>

namespace {

constexpr int kMaxSize = 524288;                    // entries per level slot
constexpr int kLevelStrideEntries = kMaxSize;       // float2 entries per level

// Level resolutions / hash sizes reproduced from the reference:
// b = 2^(1/3); res_l = floor(16 * b^l); size = next_pow2(min(res^3, 2^19))
__device__ constexpr float kRes[16] = {
    16.f, 20.f, 25.f, 32.f, 40.f, 50.f, 64.f, 80.f,
    101.f, 128.f, 161.f, 203.f, 256.f, 322.f, 406.f, 512.f};
__device__ constexpr unsigned kMask[16] = {
    4095u, 8191u, 16383u, 32767u, 65535u, 131071u, 262143u, 524287u,
    524287u, 524287u, 524287u, 524287u, 524287u, 524287u, 524287u, 524287u};

__device__ __forceinline__ void hash_corners(float res, unsigned mask,
                                             float px, float py, float pz,
                                             unsigned (&idx)[8], float (&w)[8]) {
  const float xs = px * res, ys = py * res, zs = pz * res;
  const float xf = floorf(xs), yf = floorf(ys), zf = floorf(zs);
  const float tx = xs - xf, ty = ys - yf, tz = zs - zf;
  const unsigned ix = (unsigned)xf, iy = (unsigned)yf, iz = (unsigned)zf;
  // primes: x->1, y->2654435761, z->805459861 ; (i+1)*P == i*P + P (mod 2^32)
  const unsigned hx0 = ix, hx1 = ix + 1u;
  const unsigned hy0 = iy * 2654435761u, hy1 = hy0 + 2654435761u;
  const unsigned hz0 = iz * 805459861u,  hz1 = hz0 + 805459861u;
  idx[0] = (hx0 ^ hy0 ^ hz0) & mask;
  idx[1] = (hx1 ^ hy0 ^ hz0) & mask;
  idx[2] = (hx0 ^ hy1 ^ hz0) & mask;
  idx[3] = (hx1 ^ hy1 ^ hz0) & mask;
  idx[4] = (hx0 ^ hy0 ^ hz1) & mask;
  idx[5] = (hx1 ^ hy0 ^ hz1) & mask;
  idx[6] = (hx0 ^ hy1 ^ hz1) & mask;
  idx[7] = (hx1 ^ hy1 ^ hz1) & mask;
  const float wx0 = 1.0f - tx, wy0 = 1.0f - ty, wz0 = 1.0f - tz;
  w[0] = (wx0 * wy0) * wz0;
  w[1] = (tx  * wy0) * wz0;
  w[2] = (wx0 * ty ) * wz0;
  w[3] = (tx  * ty ) * wz0;
  w[4] = (wx0 * wy0) * tz;
  w[5] = (tx  * wy0) * tz;
  w[6] = (wx0 * ty ) * tz;
  w[7] = (tx  * ty ) * tz;
}

// One level: 8 gathers batched first (loads pipeline), then the reduction.
// Works for both global and LDS tables; when `t` derives from a __shared__
// array, InferAddressSpaces turns the loads into ds_load_b64.
__device__ __forceinline__ float2 enc_level(const float2* __restrict__ t,
                                            float res, unsigned mask,
                                            float px, float py, float pz) {
  unsigned idx[8];
  float w[8];
  hash_corners(res, mask, px, py, pz, idx, w);
  const float2 v0 = t[idx[0]], v1 = t[idx[1]], v2 = t[idx[2]], v3 = t[idx[3]];
  const float2 v4 = t[idx[4]], v5 = t[idx[5]], v6 = t[idx[6]], v7 = t[idx[7]];
  float o0 = w[0] * v0.x, o1 = w[0] * v0.y;
  o0 = fmaf(w[1], v1.x, o0); o1 = fmaf(w[1], v1.y, o1);
  o0 = fmaf(w[2], v2.x, o0); o1 = fmaf(w[2], v2.y, o1);
  o0 = fmaf(w[3], v3.x, o0); o1 = fmaf(w[3], v3.y, o1);
  o0 = fmaf(w[4], v4.x, o0); o1 = fmaf(w[4], v4.y, o1);
  o0 = fmaf(w[5], v5.x, o0); o1 = fmaf(w[5], v5.y, o1);
  o0 = fmaf(w[6], v6.x, o0); o1 = fmaf(w[6], v6.y, o1);
  o0 = fmaf(w[7], v7.x, o0); o1 = fmaf(w[7], v7.y, o1);
  return make_float2(o0, o1);
}

} // namespace

__global__ __launch_bounds__(256) void MultiResHashGrid_33397665693997_kernel(
    const float* __restrict__ x, const float* __restrict__ tables,
    float* __restrict__ out, int n_pts) {
  // Levels 0 (4096 entries) and 1 (8192 entries) staged in LDS: 96 KB of the
  // 320 KB/WGP pool -> 3 concurrent workgroups (24 wave32s) per WGP.
  __shared__ float2 sh[12288];

  const int tid = threadIdx.x;

  // ---- Issue async global->LDS copies of levels 0+1 (ASYNCcnt-tracked). ----
  // 6144 16-byte chunks partitioned across the 256 threads (24 per thread);
  // overlapped with the 14 L2-resident levels below, waited on at the end.
  {
    const uint32_t lds0 = (uint32_t)(uintptr_t)(void*)&sh[0];
    const char* g0 = (const char*)tables;                                   // level 0
    const char* g1 = (const char*)(tables + (size_t)kLevelStrideEntries * 2); // level 1
#pragma unroll
    for (int i = 0; i < 8; ++i) {                       // 4096*8B = 32 KB
      const uint32_t lds = lds0 + (uint32_t)(tid + i * 256) * 16u;
      const void* g = g0 + (size_t)(tid + i * 256) * 16u;
      asm volatile("global_load_async_to_lds_b128 %0, %1, off"
                   :: "v"(lds), "v"(g) : "memory");
    }
#pragma unroll
    for (int i = 0; i < 16; ++i) {                      // 8192*8B = 64 KB
      const uint32_t lds = lds0 + 32768u + (uint32_t)(tid + i * 256) * 16u;
      const void* g = g1 + (size_t)(tid + i * 256) * 16u;
      asm volatile("global_load_async_to_lds_b128 %0, %1, off"
                   :: "v"(lds), "v"(g) : "memory");
    }
  }

  const int n = blockIdx.x * 256 + tid;
  const bool active = n < n_pts;
  float px = 0.f, py = 0.f, pz = 0.f;
  float* orow = out + (size_t)n * 32;
  const float2* tab = (const float2*)tables;

  if (active) {
    px = x[n * 3 + 0];
    py = x[n * 3 + 1];
    pz = x[n * 3 + 2];
    // Levels 2..15: hash-gather straight from global; the whole 64 MB table
    // set is L2-resident (192 MB L2), so these are L2-bandwidth bound.
#pragma unroll
    for (int lp = 1; lp < 8; ++lp) {
      const int l0 = 2 * lp, l1 = 2 * lp + 1;
      const float2 a = enc_level(tab + (size_t)l0 * kLevelStrideEntries,
                                 kRes[l0], kMask[l0], px, py, pz);
      const float2 b = enc_level(tab + (size_t)l1 * kLevelStrideEntries,
                                 kRes[l1], kMask[l1], px, py, pz);
      *reinterpret_cast<float4*>(orow + 4 * lp) = make_float4(a.x, a.y, b.x, b.y);
    }
  }

  // All of this wave's async copies done, then all waves' copies visible.
  asm volatile("s_wait_asynccnt 0" ::: "memory");
  __syncthreads();

  if (active) {
    // Levels 0,1 served from LDS (ds_load_b64 gathers).
    const float2 a = enc_level(sh,        kRes[0], kMask[0], px, py, pz);
    const float2 b = enc_level(sh + 4096, kRes[1], kMask[1], px, py, pz);
    *reinterpret_cast<float4*>(orow) = make_float4(a.x, a.y, b.x, b.y);
  }
}

extern "C" void kernel_launch(void* const* d_in, const int* in_sizes, int n_in,
                              void* d_out, int out_size, void* d_ws, size_t ws_size,
                              hipStream_t stream) {
  const float* x = (const float*)d_in[0];       // [N,3] f32
  const float* tables = (const float*)d_in[1];  // [16, 524288, 2] f32
  float* out = (float*)d_out;                   // [N,32] f32
  const int n = in_sizes[0] / 3;
  const int blocks = (n + 255) / 256;
  hipLaunchKernelGGL(MultiResHashGrid_33397665693997_kernel,
                     dim3(blocks), dim3(256), 0, stream, x, tables, out, n);
}